// SRMNeuron_3642132267482
// MI455X (gfx1250) — compile-verified
//
#include <hip/hip_runtime.h>
#include <hip/hip_bf16.h>
#include <math.h>

// SRM neuron: per-row FIR (K=50, 'same') + refractory scan.
// B=4096 rows, T=16384 steps. Bandwidth-bound (~512MB @ 23.3TB/s ~= 22us).
// Conv = banded GEMM on the WMMA pipe (17x v_wmma_f32_16x16x4_f32 per tile).
// Scan = segment-parallel 4-state automaton (ref in {0,1,2,3}).
// Staging = async global->LDS (ASYNCcnt path) when the builtin is available.

#define BDIM   4096
#define TDIM   16384
#define KLEN   50
#define ROWS   16           // rows per workgroup (= WMMA N)
#define TT     256          // time-tile per iteration
#define NCHUNK 17           // 68-wide band / K=4
#define XWIN   308          // TT + 25 left halo + 27 right halo (incl. pad)
#define XPITCH 309          // odd -> conflict-free LDS for the WMMA B reads
#define FPITCH 268          // mult of 4 (float4 align); 12*r mod 64 all-distinct banks

#if defined(__gfx1250__) && __has_builtin(__builtin_amdgcn_global_load_async_to_lds_b32)
#define USE_ASYNC_STAGE 1
#else
#define USE_ASYNC_STAGE 0
#endif

typedef __attribute__((ext_vector_type(2))) float v2f;
typedef __attribute__((ext_vector_type(8))) float v8f;

#if USE_ASYNC_STAGE
typedef __attribute__((address_space(1))) int* gbl_i32_ptr;
typedef __attribute__((address_space(3))) int* lds_i32_ptr;
#endif

__global__ __launch_bounds__(256)
void srm_fused_kernel(const float* __restrict__ I,
                      const float* __restrict__ tau_m_p,
                      const float* __restrict__ tau_s_p,
                      const float* __restrict__ v_th_p,
                      const float* __restrict__ v_reset_p,
                      float* __restrict__ out)
{
    __shared__ float    Xs[ROWS * XPITCH];   // staged input window
    __shared__ float    Fs[ROWS * FPITCH];   // filtered current tile
    __shared__ unsigned Tmap[256];           // per (row,seg) 4->4 state map
    __shared__ unsigned Sel[256];            // per (row,seg) resolved entry state
    __shared__ float    wsh[KLEN + 2];       // normalized kernel taps

    const int tid  = threadIdx.x;
    const int lane = tid & 31;
    const int wave = tid >> 5;
    const int r0   = blockIdx.x * ROWS;

    const float tau_m   = *tau_m_p;
    const float tau_s   = *tau_s_p;
    const float v_th    = *v_th_p;
    const float v_reset = *v_reset_p;

    // --- kernel taps: w[j] = (exp(-j/tau_m) - exp(-j/tau_s)) / (sum + 1e-10)
    if (tid == 0) {
        float s = 0.f;
        for (int j = 0; j < KLEN; ++j)
            s += expf(-(float)j / tau_m) - expf(-(float)j / tau_s);
        const float inv = 1.0f / (s + 1e-10f);
        for (int j = 0; j < KLEN; ++j)
            wsh[j] = (expf(-(float)j / tau_m) - expf(-(float)j / tau_s)) * inv;
    }
    __syncthreads();

    // --- Precompute A fragments (band matrix Wb[t,u] = w[t+49-u], 16x68).
    // f32 16x4 A layout: lane L -> M = L%16; VGPR0 holds K = 2*(L>=16),
    // VGPR1 holds K = 2*(L>=16)+1 within each K=4 chunk.
    const int mrow = lane & 15;
    const int half = (lane >> 4) << 1;   // 0 or 2
    v2f afrag[NCHUNK];
    for (int c = 0; c < NCHUNK; ++c) {
        float av[2];
        #pragma unroll
        for (int q = 0; q < 2; ++q) {
            const int u = (c << 2) + half + q;
            const int j = mrow + 49 - u;          // tap index
            av[q] = (j >= 0 && j < KLEN) ? wsh[j] : 0.f;
        }
        afrag[c].x = av[0];
        afrag[c].y = av[1];
    }

    // scan decomposition: this thread owns (row sr, 16-step segment sg)
    const int sr = tid >> 4;
    const int sg = tid & 15;
    int ref_carry = 0;   // tile-to-tile refractory state (valid where sg==0)

    for (int t0 = 0; t0 < TDIM; t0 += TT) {
        // --- stage input window [t0-25, t0+282] for 16 rows (coalesced)
        for (int idx = tid; idx < ROWS * XWIN; idx += 256) {
            const int rr = idx / XWIN;
            const int i  = idx - rr * XWIN;
            const int gt = t0 - 25 + i;
#if USE_ASYNC_STAGE
            if (gt >= 0 && gt < TDIM) {
                float* gp = const_cast<float*>(&I[(size_t)(r0 + rr) * TDIM + gt]);
                __builtin_amdgcn_global_load_async_to_lds_b32(
                    (gbl_i32_ptr)gp,
                    (lds_i32_ptr)&Xs[rr * XPITCH + i],
                    /*offset=*/0, /*cpol=*/0);
            } else {
                Xs[rr * XPITCH + i] = 0.f;
            }
#else
            float v = 0.f;
            if (gt >= 0 && gt < TDIM)
                v = I[(size_t)(r0 + rr) * TDIM + gt];
            Xs[rr * XPITCH + i] = v;
#endif
        }
        // --- prefetch next tile's window (global_prefetch_b8 path)
        {
            const int nt0 = t0 + TT;
            if (nt0 < TDIM) {
                for (int idx = tid; idx < ROWS * 19; idx += 256) {
                    const int rr = idx / 19;
                    const int i  = (idx - rr * 19) << 4;
                    const int gt = nt0 - 25 + i;
                    if (gt >= 0 && gt < TDIM)
                        __builtin_prefetch(&I[(size_t)(r0 + rr) * TDIM + gt], 0, 1);
                }
            }
        }
#if USE_ASYNC_STAGE
        asm volatile("s_wait_asynccnt 0x0" ::: "memory");
#endif
        __syncthreads();

        // --- banded-GEMM convolution: each wave does 2 of the 16 sub-tiles
        for (int s = wave; s < 16; s += 8) {
            const int lt = s << 4;                // sub-tile local start time
            v8f acc = {0.f, 0.f, 0.f, 0.f, 0.f, 0.f, 0.f, 0.f};
            const int br = lane & 15;             // B fragment: N = row
            for (int c = 0; c < NCHUNK; ++c) {
                const int u0 = (c << 2) + half;
                v2f bf;
                bf.x = Xs[br * XPITCH + lt + u0];
                bf.y = Xs[br * XPITCH + lt + u0 + 1];
                acc = __builtin_amdgcn_wmma_f32_16x16x4_f32(
                        /*neg_a=*/false, afrag[c],
                        /*neg_b=*/false, bf,
                        /*c_mod=*/(short)0, acc,
                        /*reuse_a=*/false, /*reuse_b=*/false);
            }
            // C layout: lane L, elem v -> M = v + 8*(L>=16), N = L%16
            const int mbase = lt + ((lane >> 4) << 3);
            #pragma unroll
            for (int v = 0; v < 8; ++v)
                Fs[br * FPITCH + mbase + v] = acc[v];
        }
        __syncthreads();

        // --- segment scan: 16-bit crossing mask (state-independent)
        unsigned crossm = 0;
        {
            const float4* fp = (const float4*)&Fs[sr * FPITCH + (sg << 4)];
            #pragma unroll
            for (int k4 = 0; k4 < 4; ++k4) {
                const float4 f4 = fp[k4];
                crossm |= ((v_reset + f4.x * 0.01f >= v_th) ? 1u : 0u) << (4*k4 + 0);
                crossm |= ((v_reset + f4.y * 0.01f >= v_th) ? 1u : 0u) << (4*k4 + 1);
                crossm |= ((v_reset + f4.z * 0.01f >= v_th) ? 1u : 0u) << (4*k4 + 2);
                crossm |= ((v_reset + f4.w * 0.01f >= v_th) ? 1u : 0u) << (4*k4 + 3);
            }
        }
        if (t0 == 0 && sg == 0) crossm &= ~1u;   // t=0: no spike, no update

        // --- simulate the 16 steps for all 4 possible entry states
        unsigned outb[4], fin[4];
        #pragma unroll
        for (int s0 = 0; s0 < 4; ++s0) {
            int ref = s0;
            unsigned bits = 0;
            #pragma unroll
            for (int k = 0; k < 16; ++k) {
                const bool cross = (crossm >> k) & 1u;
                const bool inref = ref > 0;
                const bool spike = (!inref) && cross;
                bits |= (spike ? 1u : 0u) << k;
                ref = inref ? (ref - 1) : (spike ? 3 : 0);
            }
            outb[s0] = bits;
            fin[s0]  = (unsigned)ref;
        }
        Tmap[tid] = fin[0] | (fin[1] << 2) | (fin[2] << 4) | (fin[3] << 6);
        __syncthreads();

        // --- per-row compose of the 16 segment maps (one thread per row)
        if (sg == 0) {
            int st = ref_carry;
            #pragma unroll
            for (int q = 0; q < 16; ++q) {
                Sel[(sr << 4) + q] = (unsigned)st;
                const unsigned m = Tmap[(sr << 4) + q];
                st = (int)((m >> (2 * st)) & 3u);
            }
            ref_carry = st;
        }
        __syncthreads();

        // --- emit spikes: coalesced b128 stores (adjacent tids contiguous)
        {
            const unsigned bits = outb[Sel[tid]];
            float4* op = (float4*)(out + (size_t)(r0 + sr) * TDIM + t0 + (sg << 4));
            #pragma unroll
            for (int k4 = 0; k4 < 4; ++k4) {
                float4 v;
                v.x = (float)((bits >> (4*k4 + 0)) & 1u);
                v.y = (float)((bits >> (4*k4 + 1)) & 1u);
                v.z = (float)((bits >> (4*k4 + 2)) & 1u);
                v.w = (float)((bits >> (4*k4 + 3)) & 1u);
                op[k4] = v;
            }
        }
        __syncthreads();
    }
}

extern "C" void kernel_launch(void* const* d_in, const int* in_sizes, int n_in,
                              void* d_out, int out_size, void* d_ws, size_t ws_size,
                              hipStream_t stream) {
    (void)in_sizes; (void)n_in; (void)d_ws; (void)ws_size; (void)out_size;
    const float* I       = (const float*)d_in[0];
    const float* tau_m   = (const float*)d_in[1];
    const float* tau_s   = (const float*)d_in[2];
    const float* v_th    = (const float*)d_in[3];
    const float* v_reset = (const float*)d_in[4];
    // d_in[5] = eta (unused by the reference output)
    float* out = (float*)d_out;

    dim3 grid(BDIM / ROWS);   // 256 workgroups x 16 rows
    dim3 block(256);          // 8 wave32
    hipLaunchKernelGGL(srm_fused_kernel, grid, block, 0, stream,
                       I, tau_m, tau_s, v_th, v_reset, out);
}